// KernelGuidedLoss_60833916780584
// MI455X (gfx1250) — compile-verified
//
#include <hip/hip_runtime.h>
#include <hip/hip_bf16.h>
#include <math.h>

// Tensor geometry (B=8, C=1, S=160 per spatial dim)
#define NB     8
#define S      160
#define P      159            // slice pairs per axis
#define PLANE  (160 * 160)    // 25600 elements
#define NBINS  (3 * NB * P)   // 3816: [0..1271]=D-axis, [1272..2543]=H, [2544..3815]=W
#define DCHUNK 4              // d-planes per block -> grid = 8 * 40 = 320 blocks
#define LSTRIDE 165           // 160 DW row + 5 pad DW (TDM pad: 1 DW per 32 DW). 165%64=37 (odd)
#define PLANE_LDS (S * LSTRIDE)  // 26400 DW = 105,600 B per plane buffer

typedef __attribute__((ext_vector_type(2))) float v2f;
typedef __attribute__((ext_vector_type(8))) float v8f;
typedef __attribute__((ext_vector_type(4))) unsigned int v4u;
typedef __attribute__((ext_vector_type(4))) int v4i;
typedef __attribute__((ext_vector_type(8))) int v8i;

// LDS linear index for element (h, w) of a TDM-loaded padded plane.
__device__ __forceinline__ int lidx(int h, int w) {
  return h * LSTRIDE + w + (w >> 5);
}

// Full wave32 sum via V_WMMA_F32_16X16X4_F32 (see round-1 notes). EXEC must be all-ones.
__device__ __forceinline__ float wave_sum32(float p) {
  v2f a; a[0] = p;    a[1] = 0.0f;
  v2f b; b[0] = 1.0f; b[1] = 1.0f;
  v8f c = {};
  v8f d = __builtin_amdgcn_wmma_f32_16x16x4_f32(false, a, false, b, (short)0, c,
                                                false, false);
  float s = d[0] + d[1] + d[2] + d[3] + d[4] + d[5] + d[6] + d[7];
  s += __shfl_xor(s, 16, 32);
  return s;
}

// Issue one TDM 2D-tile load: global plane (160x160 f32, row stride 160) -> LDS at
// lds_byte_addr with padding 1 DW per 32 DW (row stride 165 DW). Wave-uniform args.
// clang-23 toolchain: 6-arg builtin (v4u, v8i, v4i, v4i, v8i, i32 cpol).
__device__ __forceinline__ void tdm_load_plane(unsigned long long gaddr,
                                               unsigned lds_byte_addr) {
  v4u g0;
  g0[0] = 1u;                              // count=1 (valid descriptor), user mode
  g0[1] = lds_byte_addr;                   // lds_addr (bytes)
  g0[2] = (unsigned)(gaddr & 0xFFFFFFFFull);        // global_addr[31:0]
  g0[3] = (unsigned)((gaddr >> 32) & 0x1FFFFFFull)  // global_addr[56:32]
        | 0x80000000u;                     // type=2 ("image") in bits 127:126

  v8i g1;
  g1[0] = (int)(0x00020000u                // data_size=2 (4 bytes)
              | 0x00100000u                // pad_enable=1
              | 0x01000000u);              // pad_interval=4 (every 32 DWORDs), pad_amount=0 (+1 DW)
  g1[1] = (int)(160u << 16);               // tensor_dim0[15:0]=160 (bits 63:48 of group1)
  g1[2] = (int)(160u << 16);               // tensor_dim1[15:0]=160 (bits 95:80)
  g1[3] = (int)(160u << 16);               // tile_dim0=160 (bits 127:112)
  g1[4] = (int)160;                        // tile_dim1=160 (bits 143:128); tile_dim2=0
  g1[5] = (int)160;                        // tensor_dim0_stride[31:0]=160
  g1[6] = 0;                               // stride hi, tensor_dim1_stride lo (unused, 2D)
  g1[7] = 0;

  v4i g2; g2[0] = 0; g2[1] = 0; g2[2] = 0; g2[3] = 0;  // dims 2-3 unused (tile_dim2=0)
  v4i g3; g3[0] = 0; g3[1] = 0; g3[2] = 0; g3[3] = 0;
  v8i g4; g4[0] = 0; g4[1] = 0; g4[2] = 0; g4[3] = 0;  // trailing group (uncharacterized): zero
  g4[4] = 0; g4[5] = 0; g4[6] = 0; g4[7] = 0;

  __builtin_amdgcn_tensor_load_to_lds(g0, g1, g2, g3, g4, 0);
}

__global__ void zero_bins_kernel(float* __restrict__ bins, int n) {
  int i = blockIdx.x * blockDim.x + threadIdx.x;
  if (i < n) bins[i] = 0.0f;
}

// One block per (b, chunk of DCHUNK d-planes). 160 threads = 5 wave32.
// TDM double-buffers planes in LDS; W/H diff phases overlap the next plane's DMA.
__global__ __launch_bounds__(160) void plane_kernel(const float* __restrict__ x,
                                                    float* __restrict__ bins) {
  __shared__ float buf[2][PLANE_LDS];      // 2 x 105,600 B = 211,200 B (CDNA5: 320 KB LDS)
  __shared__ float wsum[5];

  const int t  = threadIdx.x;
  const int b  = blockIdx.x / (S / DCHUNK);
  const int d0 = (blockIdx.x % (S / DCHUNK)) * DCHUNK;
  const bool wave0   = (t < 32);
  const bool hasPair = (t < P);

  const unsigned long long xbase = (unsigned long long)(uintptr_t)x
                                 + (unsigned long long)((long)b * S + d0) * PLANE * 4ull;

  float acc_w = 0.0f;   // bin = w = t, accumulated over the whole chunk
  float acc_h = 0.0f;   // bin = h = t, accumulated over the whole chunk

  // Preload plane d0 into buf[0].
  if (wave0) {
    tdm_load_plane(xbase, (unsigned)(uintptr_t)&buf[0][0]);
    __builtin_amdgcn_s_wait_tensorcnt(0);
  }
  __syncthreads();

  int cur = 0;
  for (int j = 0; j < DCHUNK; ++j) {
    const int d   = d0 + j;
    const int nxt = cur ^ 1;
    const bool hasNext = (d + 1 < S);

    // Kick off DMA of plane d+1 while we compute on plane d.
    if (hasNext && wave0) {
      tdm_load_plane(xbase + (unsigned long long)(j + 1) * PLANE * 4ull,
                     (unsigned)(uintptr_t)&buf[nxt][0]);
    }

    const float* __restrict__ Bc = &buf[cur][0];

    // W-diffs on plane d: thread = column w = t, loop over rows (conflict-free).
    if (hasPair) {
      for (int h = 0; h < S; ++h) {
        float dw = Bc[lidx(h, t + 1)] - Bc[lidx(h, t)];
        acc_w += dw * dw;
      }
    }
    // H-diffs on plane d: thread = row h = t, transposed walk (stride 165 -> no conflicts).
    if (hasPair) {
      for (int w = 0; w < S; ++w) {
        float dh = Bc[lidx(t + 1, w)] - Bc[lidx(t, w)];
        acc_h += dh * dh;
      }
    }

    if (hasNext) {
      // Land plane d+1, publish to all waves.
      if (wave0) __builtin_amdgcn_s_wait_tensorcnt(0);
      __syncthreads();

      // D-diffs between planes d and d+1: thread = column t, loop rows.
      const float* __restrict__ Bn = &buf[nxt][0];
      float acc_d = 0.0f;
      for (int h = 0; h < S; ++h) {
        float dd = Bn[lidx(h, t)] - Bc[lidx(h, t)];
        acc_d += dd * dd;
      }
      // Block-exclusive bin: WMMA wave reduce + LDS combine, plain store.
      float ws = wave_sum32(acc_d);
      if ((t & 31) == 0) wsum[t >> 5] = ws;
      __syncthreads();
      if (t == 0) bins[b * P + d] = wsum[0] + wsum[1] + wsum[2] + wsum[3] + wsum[4];
      __syncthreads();  // buf[cur] & wsum safe to reuse after this
    }
    cur = nxt;
  }

  // Scatter chunk-accumulated W/H bins.
  if (hasPair) {
    unsafeAtomicAdd(&bins[1 * NB * P + b * P + t], acc_h);  // H axis, sigma[1]
    unsafeAtomicAdd(&bins[2 * NB * P + b * P + t], acc_w);  // W axis, sigma[2]
  }
}

// Single-wave epilogue: RBF transform per axis, WMMA reduce, scalar out.
__global__ __launch_bounds__(32) void finish_kernel(const float* __restrict__ bins,
                                                    const float* __restrict__ log_sigma,
                                                    float* __restrict__ out) {
  const int lane = threadIdx.x;
  float inv0 = 0.5f * expf(-2.0f * log_sigma[0]);  // 1/(2*sigma^2)
  float inv1 = 0.5f * expf(-2.0f * log_sigma[1]);
  float inv2 = 0.5f * expf(-2.0f * log_sigma[2]);

  float p = 0.0f;
  for (int i = lane; i < NBINS; i += 32) {
    int axis = i / (NB * P);
    float inv = (axis == 0) ? inv0 : ((axis == 1) ? inv1 : inv2);
    p += expf(-bins[i] * inv);
  }
  float tot = wave_sum32(p);
  if (lane == 0) out[0] = -tot / (3.0f * (float)(NB * P));
}

extern "C" void kernel_launch(void* const* d_in, const int* in_sizes, int n_in,
                              void* d_out, int out_size, void* d_ws, size_t ws_size,
                              hipStream_t stream) {
  const float* x         = (const float*)d_in[0];
  const float* log_sigma = (const float*)d_in[1];
  float*       out       = (float*)d_out;
  float*       bins      = (float*)d_ws;  // 3816 floats

  zero_bins_kernel<<<(NBINS + 255) / 256, 256, 0, stream>>>(bins, NBINS);
  plane_kernel<<<NB * (S / DCHUNK), 160, 0, stream>>>(x, bins);
  finish_kernel<<<1, 32, 0, stream>>>(bins, log_sigma, out);
}